// ScalarP1FunctionSpace_24232205484054
// MI455X (gfx1250) — compile-verified
//
#include <hip/hip_runtime.h>
#include <cstdint>

typedef float f4 __attribute__((ext_vector_type(4)));
typedef float f2 __attribute__((ext_vector_type(2)));
typedef int   i4 __attribute__((ext_vector_type(4)));

#define TPB   256
#define PPT   4
#define TILE  (TPB * PPT)   // 1024 points per block

__global__ __launch_bounds__(TPB)
void p1eval_kernel(const float* __restrict__ x,
                   const float* __restrict__ Minv,
                   const float* __restrict__ A,
                   const float* __restrict__ weight,
                   const int*   __restrict__ cell_ids,
                   const int*   __restrict__ dofs,
                   float*       __restrict__ out)
{
    // x-slice staging buffer: TILE points * 2 floats = 2048 f32 = 512 x b128 (8 KB).
    // Thread t owns slots 2t and 2t+1 (writes AND reads them), so the staging is
    // wave-self-contained: per-wave s_wait_asynccnt is enough, no block barrier.
    __shared__ f4 sx[TILE / 2];

    const int  t    = threadIdx.x;
    const long base = (long)blockIdx.x * TILE;     // first point of this block

    // ---- 1) Kick off CDNA5 async copy of this thread's x pair into LDS ----
    // 2 x b128 async loads per lane, tracked on ASYNCcnt, overlapping with the
    // dependent gather chain below (which lives on LOADcnt).
    {
        const f4* gx = (const f4*)(x + base * 2);
        #pragma unroll
        for (int k = 0; k < 2; ++k) {
            uint32_t lds_addr = (uint32_t)(uintptr_t)(&sx[2 * t + k]);
            uint64_t gaddr    = (uint64_t)(uintptr_t)(gx + 2 * t + k);
            asm volatile("global_load_async_to_lds_b128 %0, %1, off"
                         :: "v"(lds_addr), "v"(gaddr) : "memory");
        }
    }

    // ---- 2) Random gathers while the async copy is in flight ----
    // cell_ids is streaming (read once) -> non-temporal b128 load.
    const i4 cid = __builtin_nontemporal_load((const i4*)(cell_ids + base) + t);

    float c0[PPT], dc1[PPT], dc2[PPT];
    f4 M[PPT];
    f2 Av[PPT];
    #pragma unroll
    for (int j = 0; j < PPT; ++j) {
        const long c = (long)cid[j];
        M[j]  = *(const f4*)(Minv + 4 * c);    // [M00 M01 M10 M11], one b128
        Av[j] = *(const f2*)(A + 2 * c);       // one b64
        const int d0 = dofs[3 * c + 0];        // compiler merges -> one b96
        const int d1 = dofs[3 * c + 1];
        const int d2 = dofs[3 * c + 2];
        const float w0 = weight[d0];
        c0[j]  = w0;
        dc1[j] = weight[d1] - w0;
        dc2[j] = weight[d2] - w0;
    }

    // ---- 3) Wait for this wave's async tile, finish in registers ----
    asm volatile("s_wait_asynccnt 0x0" ::: "memory");

    f4 r;
    #pragma unroll
    for (int j = 0; j < PPT; ++j) {
        const f4  xv = sx[2 * t + (j >> 1)];   // 2 points per f4, thread-owned slots
        const float px = (j & 1) ? xv.z : xv.x;
        const float py = (j & 1) ? xv.w : xv.y;
        const float dx = px - Av[j].x;
        const float dy = py - Av[j].y;
        // st = (p - A) @ Minv  (row-major 2x2):
        //   s = dx*M00 + dy*M10 ; t = dx*M01 + dy*M11
        const float s  = fmaf(dy, M[j].z, dx * M[j].x);
        const float tt = fmaf(dy, M[j].w, dx * M[j].y);
        // c0*(1-s-t) + c1*s + c2*t = c0 + s*(c1-c0) + t*(c2-c0)
        ((float*)&r)[j] = fmaf(tt, dc2[j], fmaf(s, dc1[j], c0[j]));
    }
    // out is streaming -> non-temporal b128 store.
    __builtin_nontemporal_store(r, (f4*)(out + base) + t);
}

extern "C" void kernel_launch(void* const* d_in, const int* in_sizes, int n_in,
                              void* d_out, int out_size, void* d_ws, size_t ws_size,
                              hipStream_t stream) {
    const float* x        = (const float*)d_in[0];   // (1, N_PTS, 2)
    const float* Minv     = (const float*)d_in[1];   // (N_CELLS, 2, 2)
    const float* A        = (const float*)d_in[2];   // (N_CELLS, 2)
    const float* weight   = (const float*)d_in[3];   // (N_DOFS,)
    const int*   cell_ids = (const int*)d_in[4];     // (N_PTS,)
    const int*   dofs     = (const int*)d_in[5];     // (N_CELLS, 3)
    float*       out      = (float*)d_out;

    const int n_pts  = in_sizes[4];                  // 4194304, multiple of TILE
    const int n_blks = n_pts / TILE;                 // 4096

    p1eval_kernel<<<n_blks, TPB, 0, stream>>>(x, Minv, A, weight, cell_ids, dofs, out);
}